// MambaContrast_78743930405422
// MI455X (gfx1250) — compile-verified
//
#include <hip/hip_runtime.h>
#include <hip/hip_bf16.h>

typedef float v2f __attribute__((ext_vector_type(2)));
typedef float v8f __attribute__((ext_vector_type(8)));

// ---------------------------------------------------------------------------
// f32 WMMA: D(16x16) = A(16x4) * B(4x16) + C   -- v_wmma_f32_16x16x4_f32
// ---------------------------------------------------------------------------
__device__ __forceinline__ v8f wmma_f32(v2f a, v2f b, v8f c) {
  return __builtin_amdgcn_wmma_f32_16x16x4_f32(false, a, false, b, (short)0, c,
                                               false, false);
}

__device__ __forceinline__ float siluf(float v) { return v / (1.f + expf(-v)); }

// compile-time-vectorized pair load {p[0], p[kstride]}
template <bool VEC>
__device__ __forceinline__ v2f ld2p(const float* __restrict__ p, long long kstride) {
  v2f r;
  if (VEC) {  // kstride==1 and 8B-aligned guaranteed by caller
    const float2 t = *reinterpret_cast<const float2*>(p);
    r[0] = t.x;
    r[1] = t.y;
  } else {
    r[0] = p[0];
    r[1] = p[kstride];
  }
  return r;
}

// ---------------------------------------------------------------------------
// Generic WMMA GEMM, 16(M) x 64(N) tile per wave, software-pipelined:
//   C[m,n] = sum_k A[m*lam + k*lak] * B[n*lbn + k*lbk]  (+ biasM[m])
// ---------------------------------------------------------------------------
template <bool VECA, bool VECB>
__global__ void k_gemm_wmma(const float* __restrict__ A, const float* __restrict__ Bm,
                            const float* __restrict__ biasM, float* __restrict__ C,
                            int M, int N, int K,
                            long long lam, long long lak,
                            long long lbn, long long lbk,
                            long long lcm, long long lcn,
                            long long abs_, long long bbs, long long cbs) {
  A += (long long)blockIdx.z * abs_;
  Bm += (long long)blockIdx.z * bbs;
  C += (long long)blockIdx.z * cbs;

  const int lane = threadIdx.x & 31;
  const int wave = threadIdx.x >> 5;
  const int half = lane >> 4;
  const int l = lane & 15;

  const int ntile = (blockIdx.x * (blockDim.x >> 5) + wave) * 64;
  const int mtile = blockIdx.y * 16;
  if (ntile >= N || mtile >= M) return;  // wave-uniform

  const int m = mtile + l;
  const int K4 = K & ~3;
  v8f acc[4] = {};

  float bias8[8];
#pragma unroll
  for (int r = 0; r < 8; ++r) {
    const int mm = mtile + r + half * 8;
    bias8[r] = (biasM && mm < M) ? biasM[mm] : 0.f;
  }

  if (mtile + 16 <= M && ntile + 64 <= N) {
    // -------- interior fast path: pipelined unconditional loads --------------
    const float* pa = A + (long long)m * lam + (long long)(2 * half) * lak;
    const float* pb[4];
#pragma unroll
    for (int s = 0; s < 4; ++s)
      pb[s] = Bm + (long long)(ntile + s * 16 + l) * lbn +
              (long long)(2 * half) * lbk;
    const long long stepA = 4 * lak, stepB = 4 * lbk;
    if (K4 > 0) {
      // prologue: load first fragments
      v2f av = ld2p<VECA>(pa, lak);
      v2f bv[4];
#pragma unroll
      for (int s = 0; s < 4; ++s) bv[s] = ld2p<VECB>(pb[s], lbk);
      pa += stepA;
#pragma unroll
      for (int s = 0; s < 4; ++s) pb[s] += stepB;
      // steady state: issue next-iteration loads before consuming current
      for (int k0 = 4; k0 < K4; k0 += 4) {
        const v2f av_n = ld2p<VECA>(pa, lak);
        v2f bv_n[4];
#pragma unroll
        for (int s = 0; s < 4; ++s) bv_n[s] = ld2p<VECB>(pb[s], lbk);
        pa += stepA;
#pragma unroll
        for (int s = 0; s < 4; ++s) pb[s] += stepB;
#pragma unroll
        for (int s = 0; s < 4; ++s) acc[s] = wmma_f32(av, bv[s], acc[s]);
        av = av_n;
#pragma unroll
        for (int s = 0; s < 4; ++s) bv[s] = bv_n[s];
      }
      // epilogue
#pragma unroll
      for (int s = 0; s < 4; ++s) acc[s] = wmma_f32(av, bv[s], acc[s]);
    }
    if (K4 < K) {  // K tail: branchless k-only guards
      const int ka = K4 + 2 * half;
      v2f av;
#pragma unroll
      for (int j = 0; j < 2; ++j) {
        const int k = ka + j;
        const bool ok = (k < K);
        const long long ia = ok ? ((long long)m * lam + (long long)k * lak) : 0;
        const float t = A[ia];
        av[j] = ok ? t : 0.f;
      }
#pragma unroll
      for (int s = 0; s < 4; ++s) {
        const int n = ntile + s * 16 + l;
        v2f bv;
#pragma unroll
        for (int j = 0; j < 2; ++j) {
          const int k = ka + j;
          const bool ok = (k < K);
          const long long ib = ok ? ((long long)n * lbn + (long long)k * lbk) : 0;
          const float t = Bm[ib];
          bv[j] = ok ? t : 0.f;
        }
        acc[s] = wmma_f32(av, bv, acc[s]);
      }
    }
#pragma unroll
    for (int s = 0; s < 4; ++s) {
      const long long ncol = (long long)(ntile + s * 16 + l) * lcn;
#pragma unroll
      for (int r = 0; r < 8; ++r) {
        const int mm = mtile + r + half * 8;
        C[(long long)mm * lcm + ncol] = acc[s][r] + bias8[r];
      }
    }
  } else {
    // -------- edge path: branchless clamped-address selects ------------------
    for (int k0 = 0; k0 < K; k0 += 4) {
      const int ka = k0 + 2 * half;
      v2f av;
#pragma unroll
      for (int j = 0; j < 2; ++j) {
        const int k = ka + j;
        const bool ok = (m < M) && (k < K);
        const long long ia = ok ? ((long long)m * lam + (long long)k * lak) : 0;
        const float t = A[ia];
        av[j] = ok ? t : 0.f;
      }
#pragma unroll
      for (int s = 0; s < 4; ++s) {
        const int n = ntile + s * 16 + l;
        v2f bv;
#pragma unroll
        for (int j = 0; j < 2; ++j) {
          const int k = ka + j;
          const bool ok = (n < N) && (k < K);
          const long long ib = ok ? ((long long)n * lbn + (long long)k * lbk) : 0;
          const float t = Bm[ib];
          bv[j] = ok ? t : 0.f;
        }
        acc[s] = wmma_f32(av, bv, acc[s]);
      }
    }
#pragma unroll
    for (int s = 0; s < 4; ++s) {
#pragma unroll
      for (int r = 0; r < 8; ++r) {
        const int mm = mtile + r + half * 8;
        const int nn = ntile + s * 16 + l;
        if (mm < M && nn < N)
          C[(long long)mm * lcm + (long long)nn * lcn] = acc[s][r] + bias8[r];
      }
    }
  }
}

// ---------------------------------------------------------------------------
// 3x3 conv (pad=1, stride 1 or 2) as implicit GEMM with f32 WMMA, pipelined.
// 16(Co) x 64(pixels) per wave; Co must be a multiple of 16 (true for all calls).
// ---------------------------------------------------------------------------
template <bool VECA>
__global__ void k_conv3x3_wmma(const float* __restrict__ x, const float* __restrict__ w,
                               const float* __restrict__ bias, float* __restrict__ out,
                               int B, int Ci, int H, int W, int Co, int stride,
                               int Ho, int Wo) {
  const int lane = threadIdx.x & 31;
  const int wave = threadIdx.x >> 5;
  const int half = lane >> 4;
  const int l = lane & 15;

  const int NP = B * Ho * Wo;
  const int ptile = (blockIdx.x * (blockDim.x >> 5) + wave) * 64;
  const int co0 = blockIdx.y * 16;
  if (ptile >= NP) return;  // wave-uniform

  const int K = Ci * 9;
  const int K4 = K & ~3;
  const long long HWin = (long long)H * W;
  const int m = co0 + l;  // always < Co (Co % 16 == 0)

  long long pbase[4];
  int ih0[4], iw0[4];
  bool pv[4];
#pragma unroll
  for (int s = 0; s < 4; ++s) {
    const int p = ptile + s * 16 + l;
    pv[s] = (p < NP);
    const int pc = pv[s] ? p : 0;
    const int b = pc / (Ho * Wo);
    const int rr = pc % (Ho * Wo);
    const int oh = rr / Wo, ow = rr % Wo;
    ih0[s] = oh * stride - 1;
    iw0[s] = ow * stride - 1;
    pbase[s] = (long long)b * Ci * HWin + (long long)ih0[s] * W + iw0[s];
  }

  float bias8[8];
#pragma unroll
  for (int r = 0; r < 8; ++r)
    bias8[r] = bias ? bias[co0 + r + half * 8] : 0.f;

  // loads the 4 B fragments for k-step k0 (uniform scalar decomposition)
  auto loadB = [&](int k0, v2f bvv[4]) {
    int khv[4], kwv[4];
    long long kov[4];
#pragma unroll
    for (int i = 0; i < 4; ++i) {
      const int k = k0 + i;
      const int ci = k / 9;
      const int rr = k - ci * 9;
      const int kh = rr / 3;
      const int kw = rr - kh * 3;
      khv[i] = kh;
      kwv[i] = kw;
      kov[i] = ci * HWin + (long long)kh * W + kw;
    }
    int khj[2], kwj[2];
    long long koj[2];
#pragma unroll
    for (int j = 0; j < 2; ++j) {
      khj[j] = half ? khv[2 + j] : khv[j];
      kwj[j] = half ? kwv[2 + j] : kwv[j];
      koj[j] = half ? kov[2 + j] : kov[j];
    }
#pragma unroll
    for (int s = 0; s < 4; ++s) {
#pragma unroll
      for (int j = 0; j < 2; ++j) {
        const int ih = ih0[s] + khj[j];
        const int iw = iw0[s] + kwj[j];
        const bool ok = pv[s] && ((unsigned)ih < (unsigned)H) &&
                        ((unsigned)iw < (unsigned)W);
        const long long ib = ok ? (pbase[s] + koj[j]) : 0;
        const float t = x[ib];
        bvv[s][j] = ok ? t : 0.f;
      }
    }
  };

  v8f acc[4] = {};
  const float* pa = w + (long long)m * K + 2 * half;
  if (K4 > 0) {
    // prologue
    v2f av = ld2p<VECA>(pa, 1);
    v2f bv[4];
    loadB(0, bv);
    pa += 4;
    // steady state: issue next loads before consuming current fragments
    for (int k0 = 4; k0 < K4; k0 += 4) {
      const v2f av_n = ld2p<VECA>(pa, 1);
      v2f bv_n[4];
      loadB(k0, bv_n);
      pa += 4;
#pragma unroll
      for (int s = 0; s < 4; ++s) acc[s] = wmma_f32(av, bv[s], acc[s]);
      av = av_n;
#pragma unroll
      for (int s = 0; s < 4; ++s) bv[s] = bv_n[s];
    }
    // epilogue
#pragma unroll
    for (int s = 0; s < 4; ++s) acc[s] = wmma_f32(av, bv[s], acc[s]);
  }
  if (K4 < K) {  // K tail (e.g. Ci=3 -> K=27)
    const int ka = K4 + 2 * half;
    v2f av;
#pragma unroll
    for (int j = 0; j < 2; ++j) {
      const int k = ka + j;
      const bool ok = (k < K);
      const long long ia = ok ? ((long long)m * K + k) : 0;
      const float t = w[ia];
      av[j] = ok ? t : 0.f;
    }
#pragma unroll
    for (int s = 0; s < 4; ++s) {
      v2f bv;
#pragma unroll
      for (int j = 0; j < 2; ++j) {
        const int k = ka + j;
        const int ci = k / 9;
        const int rr = k - ci * 9;
        const int kh = rr / 3;
        const int kw = rr - kh * 3;
        const int ih = ih0[s] + kh;
        const int iw = iw0[s] + kw;
        const bool ok = (k < K) && pv[s] && ((unsigned)ih < (unsigned)H) &&
                        ((unsigned)iw < (unsigned)W);
        const long long ib =
            ok ? (pbase[s] + ci * HWin + (long long)kh * W + kw) : 0;
        const float t = x[ib];
        bv[j] = ok ? t : 0.f;
      }
      acc[s] = wmma_f32(av, bv, acc[s]);
    }
  }
  const long long HWo = (long long)Ho * Wo;
#pragma unroll
  for (int s = 0; s < 4; ++s) {
    const int pp = ptile + s * 16 + l;
    const int pc = (pp < NP) ? pp : 0;
    const int bb = pc / (Ho * Wo);
    const int rr = pc % (Ho * Wo);
    const long long obase = (long long)bb * Co * HWo + rr;
#pragma unroll
    for (int r = 0; r < 8; ++r) {
      const int mm = co0 + r + half * 8;
      if (pp < NP) out[obase + (long long)mm * HWo] = acc[s][r] + bias8[r];
    }
  }
}

// ---------------------------------------------------------------------------
// GroupNorm + SiLU (one block per (batch, group))
// ---------------------------------------------------------------------------
__global__ void k_gn_silu(const float* __restrict__ x, const float* __restrict__ g,
                          const float* __restrict__ bt, float* __restrict__ o,
                          int C, long long HW, int groups) {
  const int b = blockIdx.x / groups;
  const int grp = blockIdx.x % groups;
  const int cpg = C / groups;
  const long long base = ((long long)b * C + (long long)grp * cpg) * HW;
  const long long n = (long long)cpg * HW;
  float s = 0.f, s2 = 0.f;
  for (long long i = threadIdx.x; i < n; i += blockDim.x) {
    float v = x[base + i];
    s += v;
    s2 += v * v;
  }
  for (int off = 16; off; off >>= 1) {
    s += __shfl_xor(s, off, 32);
    s2 += __shfl_xor(s2, off, 32);
  }
  __shared__ float sh0[16], sh1[16];
  const int wv = threadIdx.x >> 5;
  if ((threadIdx.x & 31) == 0) { sh0[wv] = s; sh1[wv] = s2; }
  __syncthreads();
  if (threadIdx.x == 0) {
    float a = 0.f, c2 = 0.f;
    const int nw = (blockDim.x + 31) >> 5;
    for (int i = 0; i < nw; ++i) { a += sh0[i]; c2 += sh1[i]; }
    const float mean = a / (float)n;
    const float var = c2 / (float)n - mean * mean;
    sh0[0] = mean;
    sh1[0] = rsqrtf(var + 1e-5f);
  }
  __syncthreads();
  const float mean = sh0[0], inv = sh1[0];
  for (long long i = threadIdx.x; i < n; i += blockDim.x) {
    const int ci = grp * cpg + (int)(i / HW);
    float v = (x[base + i] - mean) * inv * g[ci] + bt[ci];
    o[base + i] = siluf(v);
  }
}

// ---------------------------------------------------------------------------
// BatchNorm (batch stats over N,H,W) + ReLU (one block per channel)
// ---------------------------------------------------------------------------
__global__ void k_bn_relu(const float* __restrict__ x, const float* __restrict__ g,
                          const float* __restrict__ bt, float* __restrict__ o,
                          int B, int C, long long HW) {
  const int c = blockIdx.x;
  const long long n = (long long)B * HW;
  float s = 0.f, s2 = 0.f;
  for (long long i = threadIdx.x; i < n; i += blockDim.x) {
    const long long b = i / HW, hw = i % HW;
    float v = x[(b * C + c) * HW + hw];
    s += v;
    s2 += v * v;
  }
  for (int off = 16; off; off >>= 1) {
    s += __shfl_xor(s, off, 32);
    s2 += __shfl_xor(s2, off, 32);
  }
  __shared__ float sh0[16], sh1[16];
  const int wv = threadIdx.x >> 5;
  if ((threadIdx.x & 31) == 0) { sh0[wv] = s; sh1[wv] = s2; }
  __syncthreads();
  if (threadIdx.x == 0) {
    float a = 0.f, c2 = 0.f;
    const int nw = (blockDim.x + 31) >> 5;
    for (int i = 0; i < nw; ++i) { a += sh0[i]; c2 += sh1[i]; }
    const float mean = a / (float)n;
    const float var = c2 / (float)n - mean * mean;
    sh0[0] = mean;
    sh1[0] = rsqrtf(var + 1e-5f);
  }
  __syncthreads();
  const float mean = sh0[0], inv = sh1[0], gg = g[c], bb = bt[c];
  for (long long i = threadIdx.x; i < n; i += blockDim.x) {
    const long long b = i / HW, hw = i % HW;
    const long long idx = (b * C + c) * HW + hw;
    float v = (x[idx] - mean) * inv * gg + bb;
    o[idx] = v > 0.f ? v : 0.f;
  }
}

// ---------------------------------------------------------------------------
// Mamba depthwise causal conv1d (k=4, left-pad 3) + bias + SiLU
// ---------------------------------------------------------------------------
__global__ void k_dwconv_silu(const float* __restrict__ uz, const float* __restrict__ cw,
                              const float* __restrict__ cb, float* __restrict__ u,
                              long long L, int di) {
  const long long i = blockIdx.x * (long long)blockDim.x + threadIdx.x;
  if (i >= L * di) return;
  const long long l = i / di;
  const int c = (int)(i % di);
  float acc = cb[c];
#pragma unroll
  for (int j = 0; j < 4; ++j) {
    const long long ll = l - 3 + j;
    if (ll >= 0) acc += cw[c * 4 + j] * uz[ll * (2LL * di) + c];
  }
  u[i] = siluf(acc);
}

// delta = softplus(delta_raw + dt_proj_b)
__global__ void k_softplus_bias(float* __restrict__ dlt, const float* __restrict__ dtb,
                                long long total, int di) {
  const long long i = blockIdx.x * (long long)blockDim.x + threadIdx.x;
  if (i >= total) return;
  const float v = dlt[i] + dtb[(int)(i % di)];
  dlt[i] = (v > 20.f) ? v : log1pf(expf(v));
}

// ---------------------------------------------------------------------------
// Selective scan: one lane per (d, n) state; 16-lane reduce per timestep.
// ---------------------------------------------------------------------------
__global__ void k_scan(const float* __restrict__ u, const float* __restrict__ dlt,
                       const float* __restrict__ Alog, const float* __restrict__ dbl,
                       const float* __restrict__ Dp, float* __restrict__ y,
                       long long L, int di, int rank) {
  const int t16 = blockIdx.x * blockDim.x + threadIdx.x;
  const int d = t16 >> 4;
  const int n = t16 & 15;
  if (d >= di) return;
  const float A = -expf(Alog[d * 16 + n]);
  const int RL = rank + 32;
  const float Dd = Dp[d];
  float h = 0.f;
  for (long long t = 0; t < L; ++t) {
    const float dt = dlt[t * di + d];
    const float ut = u[t * di + d];
    const float Bc = dbl[t * RL + rank + n];
    const float Cc = dbl[t * RL + rank + 16 + n];
    h = expf(dt * A) * h + dt * ut * Bc;
    float p = h * Cc;
    p += __shfl_xor(p, 1, 32);
    p += __shfl_xor(p, 2, 32);
    p += __shfl_xor(p, 4, 32);
    p += __shfl_xor(p, 8, 32);
    if (n == 0) y[t * di + d] = p + ut * Dd;
  }
}

// y *= silu(z);  z = uz[:, di:2di]
__global__ void k_gate_silu(float* __restrict__ y, const float* __restrict__ uz,
                            long long total, int di) {
  const long long i = blockIdx.x * (long long)blockDim.x + threadIdx.x;
  if (i >= total) return;
  const long long l = i / di;
  const int d = (int)(i % di);
  y[i] *= siluf(uz[l * (2LL * di) + di + d]);
}

// ---------------------------------------------------------------------------
// ConvTranspose 2x2 stride 2 (lax.conv_transpose default => kernel flipped):
// out[b,o,2h+dh,2w+dw] = b[o] + sum_i x[b,i,h,w]*w[i,o,1-dh,1-dw]
// ---------------------------------------------------------------------------
__global__ void k_convt2x2(const float* __restrict__ x, const float* __restrict__ w,
                           const float* __restrict__ bias, float* __restrict__ out,
                           int B, int Ci, int Hi, int Wi, int Co) {
  const int Ho = 2 * Hi, Wo = 2 * Wi;
  const long long total = (long long)B * Co * Ho * Wo;
  const long long idx = blockIdx.x * (long long)blockDim.x + threadIdx.x;
  if (idx >= total) return;
  const long long hwp = idx % ((long long)Ho * Wo);
  const long long t = idx / ((long long)Ho * Wo);
  const int co = (int)(t % Co);
  const int b = (int)(t / Co);
  const int oh = (int)(hwp / Wo), ow = (int)(hwp % Wo);
  const int h = oh >> 1, wq = ow >> 1;
  const int dh = 1 - (oh & 1), dw = 1 - (ow & 1);
  float acc = bias[co];
  for (int ci = 0; ci < Ci; ++ci)
    acc += x[(((long long)b * Ci + ci) * Hi + h) * Wi + wq] *
           w[(((long long)ci * Co + co) * 2 + dh) * 2 + dw];
  out[idx] = acc;
}

// softmax over batch axis (size 2): per = C*H*W
__global__ void k_softmax_b2(const float* __restrict__ s, float* __restrict__ o,
                             long long per) {
  const long long i = blockIdx.x * (long long)blockDim.x + threadIdx.x;
  if (i >= per) return;
  const float a = s[i], b = s[i + per];
  const float mx = fmaxf(a, b);
  const float ea = expf(a - mx), eb = expf(b - mx);
  const float inv = 1.f / (ea + eb);
  o[i] = ea * inv;
  o[i + per] = eb * inv;
}

// L2 normalize over channel axis, in place
__global__ void k_l2norm_c(float* __restrict__ e, int B, int C, long long HW) {
  const long long idx = blockIdx.x * (long long)blockDim.x + threadIdx.x;
  if (idx >= (long long)B * HW) return;
  const long long b = idx / HW, hw = idx % HW;
  float* p = e + b * C * HW + hw;
  float s = 0.f;
  for (int c = 0; c < C; ++c) { const float v = p[(long long)c * HW]; s += v * v; }
  const float inv = 1.f / fmaxf(sqrtf(s), 1e-12f);
  for (int c = 0; c < C; ++c) p[(long long)c * HW] *= inv;
}

// ===========================================================================
// Host-side orchestration
// ===========================================================================
static inline void gemm(hipStream_t s, const float* A, const float* B,
                        const float* biasM, float* C, int M, int N, int K,
                        long long lam, long long lak, long long lbn, long long lbk,
                        long long lcm, long long lcn, long long abs_, long long bbs,
                        long long cbs, int nb) {
  const int wpb = 4;  // waves/block; each wave owns a 16x64 tile
  dim3 grid((N + 64 * wpb - 1) / (64 * wpb), (M + 15) / 16, nb);
  const bool va = (lak == 1) && ((lam & 1) == 0);
  const bool vb = (lbk == 1) && ((lbn & 1) == 0);
  if (va && vb)
    k_gemm_wmma<true, true><<<grid, wpb * 32, 0, s>>>(A, B, biasM, C, M, N, K, lam,
                                                      lak, lbn, lbk, lcm, lcn, abs_,
                                                      bbs, cbs);
  else if (va)
    k_gemm_wmma<true, false><<<grid, wpb * 32, 0, s>>>(A, B, biasM, C, M, N, K, lam,
                                                       lak, lbn, lbk, lcm, lcn, abs_,
                                                       bbs, cbs);
  else if (vb)
    k_gemm_wmma<false, true><<<grid, wpb * 32, 0, s>>>(A, B, biasM, C, M, N, K, lam,
                                                       lak, lbn, lbk, lcm, lcn, abs_,
                                                       bbs, cbs);
  else
    k_gemm_wmma<false, false><<<grid, wpb * 32, 0, s>>>(A, B, biasM, C, M, N, K, lam,
                                                        lak, lbn, lbk, lcm, lcn, abs_,
                                                        bbs, cbs);
}

static inline void conv3x3(hipStream_t s, const float* x, const float* w,
                           const float* b, float* o, int B, int Ci, int H, int W,
                           int Co, int stride) {
  const int Ho = (H + 2 - 3) / stride + 1;
  const int Wo = (W + 2 - 3) / stride + 1;
  const int NP = B * Ho * Wo;
  const int wpb = 4;
  dim3 grid((NP + 64 * wpb - 1) / (64 * wpb), Co / 16, 1);  // Co % 16 == 0
  const int K = Ci * 9;
  if ((K & 1) == 0)
    k_conv3x3_wmma<true><<<grid, wpb * 32, 0, s>>>(x, w, b, o, B, Ci, H, W, Co,
                                                   stride, Ho, Wo);
  else
    k_conv3x3_wmma<false><<<grid, wpb * 32, 0, s>>>(x, w, b, o, B, Ci, H, W, Co,
                                                    stride, Ho, Wo);
}

static inline void conv1x1(hipStream_t s, const float* x, const float* w,
                           const float* b, float* o, int B, int Ci, long long HW,
                           int Co) {
  gemm(s, w, x, b, o, Co, (int)HW, Ci, Ci, 1, 1, HW, HW, 1, 0, (long long)Ci * HW,
       (long long)Co * HW, B);
}

static void run_mamba(hipStream_t s, const float* xin, float* xout, int B, int d_model,
                      int H, int Wd, const float* ipw, const float* cw, const float* cb,
                      const float* xpw, const float* dtw, const float* dtb,
                      const float* Alog, const float* Dp, const float* opw, float* ws) {
  const int di = 2 * d_model;
  const int rank = (d_model + 15) / 16;
  const long long HW = (long long)H * Wd;
  const long long L = (long long)B * HW;
  const int RL = rank + 32;
  float* uz = ws;
  float* u = uz + L * 2 * di;
  float* dbl = u + L * di;
  float* dlt = dbl + L * RL;
  float* y = dlt + L * di;

  // in_proj: uz(L,2di) = xf(L,d_model) @ Wip^T, reading NCHW directly per batch
  gemm(s, xin, ipw, nullptr, uz, (int)HW, 2 * di, d_model, 1, HW, d_model, 1, 2 * di,
       1, (long long)d_model * HW, 0, HW * 2 * di, B);
  {
    const long long tot = L * di;
    k_dwconv_silu<<<(unsigned)((tot + 255) / 256), 256, 0, s>>>(uz, cw, cb, u, L, di);
  }
  // x_proj: dbl(L,RL) = u @ Wxp^T
  gemm(s, u, xpw, nullptr, dbl, (int)L, RL, di, di, 1, di, 1, RL, 1, 0, 0, 0, 1);
  // dt_proj: dlt(L,di) = dbl[:, :rank] @ Wdt^T ; then +b, softplus
  gemm(s, dbl, dtw, nullptr, dlt, (int)L, di, rank, RL, 1, rank, 1, di, 1, 0, 0, 0, 1);
  {
    const long long tot = L * di;
    k_softplus_bias<<<(unsigned)((tot + 255) / 256), 256, 0, s>>>(dlt, dtb, tot, di);
  }
  {
    const int tot = di * 16;
    k_scan<<<(tot + 255) / 256, 256, 0, s>>>(u, dlt, Alog, dbl, Dp, y, L, di, rank);
  }
  {
    const long long tot = L * di;
    k_gate_silu<<<(unsigned)((tot + 255) / 256), 256, 0, s>>>(y, uz, tot, di);
  }
  // out_proj: NCHW scatter per batch
  gemm(s, y, opw, nullptr, xout, (int)HW, d_model, di, di, 1, di, 1, 1, HW,
       HW * di, 0, (long long)d_model * HW, B);
}

extern "C" void kernel_launch(void* const* d_in, const int* in_sizes, int n_in,
                              void* d_out, int out_size, void* d_ws, size_t ws_size,
                              hipStream_t stream) {
  (void)in_sizes; (void)n_in; (void)out_size; (void)ws_size;
#define P(i) ((const float*)d_in[(i)])
  const float* xs = P(0);
  float* ws = (float*)d_ws;
  float* T1 = ws;                    // 18,874,368 floats
  float* T2 = T1 + 18874368LL;       // 18,874,368
  float* T3 = T2 + 18874368LL;       //  9,437,184  (x6)
  float* MB = T3 + 9437184LL;        // mamba scratch (~26.1M)
  float* seg_out = (float*)d_out;                 // (2,9,192,192)
  float* emb = (float*)d_out + 663552LL;          // (2,256,192,192)

  const int B = 2;

  // conv1 (3->32) + GN(8) + SiLU
  conv3x3(stream, xs, P(1), P(2), T1, B, 3, 192, 192, 32, 1);
  k_gn_silu<<<B * 8, 256, 0, stream>>>(T1, P(3), P(4), T1, 32, 192LL * 192, 8);

  // mamba 1 (d_model=32): T1 -> T2
  run_mamba(stream, T1, T2, B, 32, 192, 192, P(23), P(24), P(25), P(26), P(27),
            P(28), P(29), P(30), P(31), MB);

  // down4: 32->64 s2
  conv3x3(stream, T2, P(13), P(14), T1, B, 32, 192, 192, 64, 2);

  // conv2 (64->64) + GN(16) + SiLU
  conv3x3(stream, T1, P(5), P(6), T2, B, 64, 96, 96, 64, 1);
  k_gn_silu<<<B * 16, 256, 0, stream>>>(T2, P(7), P(8), T2, 64, 96LL * 96, 16);

  // mamba 2 (d_model=64): T2 -> T1
  run_mamba(stream, T2, T1, B, 64, 96, 96, P(32), P(33), P(34), P(35), P(36),
            P(37), P(38), P(39), P(40), MB);

  // down2: 64->128 s2
  conv3x3(stream, T1, P(15), P(16), T2, B, 64, 96, 96, 128, 2);

  // conv3 (128->128) + GN(32) + SiLU
  conv3x3(stream, T2, P(9), P(10), T1, B, 128, 48, 48, 128, 1);
  k_gn_silu<<<B * 32, 256, 0, stream>>>(T1, P(11), P(12), T1, 128, 48LL * 48, 32);

  // mamba 3 (d_model=128): T1 -> T2
  run_mamba(stream, T1, T2, B, 128, 48, 48, P(41), P(42), P(43), P(44), P(45),
            P(46), P(47), P(48), P(49), MB);

  // down3: 128->256 s2
  conv3x3(stream, T2, P(17), P(18), T1, B, 128, 48, 48, 256, 2);

  // up2 / up1 / up1 (reused)
  {
    const long long tot = (long long)B * 128 * 48 * 48;
    k_convt2x2<<<(unsigned)((tot + 255) / 256), 256, 0, stream>>>(T1, P(19), P(20),
                                                                  T2, B, 256, 24, 24, 128);
  }
  {
    const long long tot = (long long)B * 128 * 96 * 96;
    k_convt2x2<<<(unsigned)((tot + 255) / 256), 256, 0, stream>>>(T2, P(21), P(22),
                                                                  T1, B, 128, 48, 48, 128);
  }
  {
    const long long tot = (long long)B * 128 * 192 * 192;
    k_convt2x2<<<(unsigned)((tot + 255) / 256), 256, 0, stream>>>(T1, P(21), P(22),
                                                                  T3, B, 128, 96, 96, 128);
  }

  // seg head
  conv3x3(stream, T3, P(50), P(51), T1, B, 128, 192, 192, 256, 1);
  k_bn_relu<<<256, 256, 0, stream>>>(T1, P(52), P(53), T1, B, 256, 192LL * 192);
  conv1x1(stream, T1, P(54), nullptr, T2, B, 256, 192LL * 192, 9);
  {
    const long long per = 9LL * 192 * 192;
    k_softmax_b2<<<(unsigned)((per + 255) / 256), 256, 0, stream>>>(T2, seg_out, per);
  }

  // projection head
  conv1x1(stream, T3, P(55), P(56), T1, B, 128, 192LL * 192, 128);
  k_bn_relu<<<128, 256, 0, stream>>>(T1, P(57), P(58), T1, B, 128, 192LL * 192);
  conv1x1(stream, T1, P(59), P(60), emb, B, 128, 192LL * 192, 256);
  {
    const long long tot = (long long)B * 192 * 192;
    k_l2norm_c<<<(unsigned)((tot + 255) / 256), 256, 0, stream>>>(emb, B, 256,
                                                                  192LL * 192);
  }
#undef P
}